// TransH_53867479826773
// MI455X (gfx1250) — compile-verified
//
#include <hip/hip_runtime.h>
#include <hip/hip_bf16.h>
#include <stdint.h>

#define EMB_D 60

// ---------------------------------------------------------------------------
// Kernel 1: convert sorted segment-id array into per-segment start offsets.
// starts[b] = first index i with seg[i] >= b ; starts[B] = T.
// Handles (unexpected) empty segments correctly for sorted seg.
// ---------------------------------------------------------------------------
__global__ void build_starts_kernel(const int* __restrict__ seg,
                                    int* __restrict__ starts,
                                    int T, int B) {
  int i = blockIdx.x * blockDim.x + threadIdx.x;
  if (i >= T) return;
  int s = seg[i];
  if (i == 0) {
    for (int b = 0; b <= s; ++b) starts[b] = 0;
  } else {
    int p = seg[i - 1];
    for (int b = p + 1; b <= s; ++b) starts[b] = i;
  }
  if (i == T - 1) {
    for (int b = s + 1; b <= B; ++b) starts[b] = T;
  }
}

// ---------------------------------------------------------------------------
// Helpers
// ---------------------------------------------------------------------------
__device__ __forceinline__ float rsum16(float v) {
  // sum across a 16-lane group (masks 1,2,4,8 never cross the 16-lane half)
  v += __shfl_xor(v, 1, 32);
  v += __shfl_xor(v, 2, 32);
  v += __shfl_xor(v, 4, 32);
  v += __shfl_xor(v, 8, 32);
  return v;
}

__device__ __forceinline__ float dot4(const float4& a, const float4& b) {
  return a.x * b.x + a.y * b.y + a.z * b.z + a.w * b.w;
}

// CDNA5 async global->LDS copy, 16 bytes per active lane. Per-lane global
// address (VGPR pair) and per-lane LDS byte offset (low 32 bits of the
// generic pointer are the LDS offset per the gfx1250 aperture rules).
__device__ __forceinline__ void async_ld16(const float* g, const float* lds) {
  uint32_t loff = (uint32_t)(uintptr_t)lds;
  uint64_t ga = (uint64_t)(uintptr_t)g;
  asm volatile("global_load_async_to_lds_b128 %0, %1, off"
               :: "v"(loff), "v"(ga)
               : "memory");
}

__device__ __forceinline__ void wait_async0() {
  asm volatile("s_wait_asynccnt 0" ::: "memory");
}

// Ragged segment-mean pooling: 16 lanes cooperate on one sample; lane j holds
// columns [4j, 4j+4). Word indices for the sample are contiguous [s, e);
// fetch up to 16 of them with one coalesced per-lane load, broadcast by shfl.
__device__ __forceinline__ float4 pool_mean(const float* __restrict__ wemb,
                                            const int* __restrict__ idx,
                                            int s, int e, int col,
                                            bool valid, int j) {
  float4 acc = make_float4(0.f, 0.f, 0.f, 0.f);
  int cnt = e - s;
  int myi = (j < cnt) ? idx[s + j] : 0;
  for (int k = 0; k < cnt; ++k) {
    int wi = (k < 16) ? __shfl(myi, k, 16) : idx[s + k];
    if (valid) {
      const float4 v =
          *reinterpret_cast<const float4*>(&wemb[(size_t)wi * EMB_D + col]);
      acc.x += v.x; acc.y += v.y; acc.z += v.z; acc.w += v.w;
    }
  }
  float c = fmaxf((float)cnt, 1.0f);
  acc.x /= c; acc.y /= c; acc.z /= c; acc.w /= c;
  return acc;
}

// ---------------------------------------------------------------------------
// Kernel 2: TransH score. 256 threads = 8 wave32 = 16 samples per block.
// ---------------------------------------------------------------------------
__global__ void __launch_bounds__(256, 1)
transh_score_kernel(const int* __restrict__ head,
                    const int* __restrict__ rel,
                    const int* __restrict__ tail,
                    const int* __restrict__ hw_idx,
                    const int* __restrict__ rw_idx,
                    const int* __restrict__ tw_idx,
                    const int* __restrict__ hstart,
                    const int* __restrict__ rstart,
                    const int* __restrict__ tstart,
                    const float* __restrict__ ent,
                    const float* __restrict__ rele,
                    const float* __restrict__ wemb,
                    const float* __restrict__ proj,
                    float* __restrict__ out,
                    int B) {
  __shared__ float smem[16 * 4 * 64];   // 16 groups x 4 rows x 64 floats (16 KB)

  const int tid = threadIdx.x;
  const int g = tid >> 4;       // group (sample slot) within block
  const int j = tid & 15;       // lane within group
  const int b = blockIdx.x * 16 + g;
  if (b >= B) return;

  const bool valid = j < 15;    // 15 lanes x float4 = 60 columns
  const int col = j * 4;

  const int h = head[b];
  const int r = rel[b];
  const int t = tail[b];

  // Stage the 4 base rows into LDS asynchronously; overlap with pooling.
  float* slot = &smem[g * 4 * 64];
  if (valid) {
    async_ld16(&ent [(size_t)h * EMB_D + col], &slot[0 * 64 + col]);
    async_ld16(&ent [(size_t)t * EMB_D + col], &slot[1 * 64 + col]);
    async_ld16(&rele[(size_t)r * EMB_D + col], &slot[2 * 64 + col]);
    async_ld16(&proj[(size_t)r * EMB_D + col], &slot[3 * 64 + col]);
  }

  const int hs = hstart[b], he = hstart[b + 1];
  const int rs = rstart[b], re = rstart[b + 1];
  const int ts = tstart[b], te = tstart[b + 1];

  float4 hm = pool_mean(wemb, hw_idx, hs, he, col, valid, j);
  float4 rm = pool_mean(wemb, rw_idx, rs, re, col, valid, j);
  float4 tm = pool_mean(wemb, tw_idx, ts, te, col, valid, j);

  wait_async0();  // staged rows now visible to this wave's LDS reads

  float4 hv = make_float4(0.f, 0.f, 0.f, 0.f);
  float4 tv = hv, rv = hv, wv = hv;
  if (valid) {
    float4 a = *reinterpret_cast<const float4*>(&slot[0 * 64 + col]);
    float4 c = *reinterpret_cast<const float4*>(&slot[1 * 64 + col]);
    float4 d = *reinterpret_cast<const float4*>(&slot[2 * 64 + col]);
    float4 e = *reinterpret_cast<const float4*>(&slot[3 * 64 + col]);
    hv.x = a.x + hm.x; hv.y = a.y + hm.y; hv.z = a.z + hm.z; hv.w = a.w + hm.w;
    tv.x = c.x + tm.x; tv.y = c.y + tm.y; tv.z = c.z + tm.z; tv.w = c.w + tm.w;
    rv.x = d.x + rm.x; rv.y = d.y + rm.y; rv.z = d.z + rm.z; rv.w = d.w + rm.w;
    wv = e;
  }

  // w = w / max(||w||, 1e-12)
  float w2 = rsum16(dot4(wv, wv));
  float denom = fmaxf(sqrtf(w2), 1e-12f);
  wv.x /= denom; wv.y /= denom; wv.z /= denom; wv.w /= denom;

  float dh = rsum16(dot4(wv, hv));
  float dt = rsum16(dot4(wv, tv));

  // d = (h - (w.h)w) + r - (t - (w.t)w)
  float4 dv;
  dv.x = (hv.x - dh * wv.x) + rv.x - (tv.x - dt * wv.x);
  dv.y = (hv.y - dh * wv.y) + rv.y - (tv.y - dt * wv.y);
  dv.z = (hv.z - dh * wv.z) + rv.z - (tv.z - dt * wv.z);
  dv.w = (hv.w - dh * wv.w) + rv.w - (tv.w - dt * wv.w);

  float s2 = rsum16(dot4(dv, dv));
  if (j == 0) out[b] = -sqrtf(s2);
}

// ---------------------------------------------------------------------------
// Launch
// ---------------------------------------------------------------------------
extern "C" void kernel_launch(void* const* d_in, const int* in_sizes, int n_in,
                              void* d_out, int out_size, void* d_ws, size_t ws_size,
                              hipStream_t stream) {
  (void)n_in; (void)out_size; (void)ws_size;

  const int*   head   = (const int*)  d_in[0];
  const int*   rel    = (const int*)  d_in[1];
  const int*   tail   = (const int*)  d_in[2];
  const int*   hw_idx = (const int*)  d_in[3];
  const int*   hw_seg = (const int*)  d_in[4];
  const int*   rw_idx = (const int*)  d_in[5];
  const int*   rw_seg = (const int*)  d_in[6];
  const int*   tw_idx = (const int*)  d_in[7];
  const int*   tw_seg = (const int*)  d_in[8];
  const float* ent    = (const float*)d_in[9];
  const float* rele   = (const float*)d_in[10];
  const float* wemb   = (const float*)d_in[11];
  const float* proj   = (const float*)d_in[12];
  float* out = (float*)d_out;

  const int B = in_sizes[0];
  const int T = in_sizes[3];

  int* hstart = (int*)d_ws;           // B+1 ints
  int* rstart = hstart + (B + 1);     // B+1 ints
  int* tstart = rstart + (B + 1);     // B+1 ints

  const int tb = 256;
  const int tg = (T + tb - 1) / tb;
  build_starts_kernel<<<tg, tb, 0, stream>>>(hw_seg, hstart, T, B);
  build_starts_kernel<<<tg, tb, 0, stream>>>(rw_seg, rstart, T, B);
  build_starts_kernel<<<tg, tb, 0, stream>>>(tw_seg, tstart, T, B);

  transh_score_kernel<<<(B + 15) / 16, 256, 0, stream>>>(
      head, rel, tail, hw_idx, rw_idx, tw_idx,
      hstart, rstart, tstart, ent, rele, wemb, proj, out, B);
}